// EdgeGCN_3453153706429
// MI455X (gfx1250) — compile-verified
//
#include <hip/hip_runtime.h>
#include <hip/hip_bf16.h>
#include <cstdint>
#include <cstddef>

// ---------------------------------------------------------------------------
// EdgeGCN for MI455X (gfx1250, wave32).
//  * Edge/node GEMMs run on V_WMMA_F32_16X16X4_F32 (exact f32).
//  * Global edge transform folded into per-layer combined weights:
//      eft_i = edge_attr @ (edge_weight@eW_i) + (edge_bias@eW_i + eb_i)
//    -> HBM traffic is just 3 streaming passes over edge_attr (~615 MB,
//       ~26 us at 23.3 TB/s).  Node tensors (12.8 MB) are L2-resident, so
//       gathers and f32 scatter-atomics stay on-chip.
//  * Epilogues use one 64-bit base address per matrix row + immediate
//    offsets for the 4 column tiles (minimizes VALU address math next to
//    the atomic/gather stream).
// ---------------------------------------------------------------------------

typedef __attribute__((ext_vector_type(2))) float v2f;
typedef __attribute__((ext_vector_type(8))) float v8f;

__device__ __forceinline__ v8f wmma4(v2f a, v2f b, v8f c) {
  // D = A(16x4) * B(4x16) + C(16x16), f32 everywhere.
  return __builtin_amdgcn_wmma_f32_16x16x4_f32(
      /*neg_a=*/false, a, /*neg_b=*/false, b,
      /*c_mod=*/(short)0, c, /*reuse_a=*/false, /*reuse_b=*/false);
}

// ---------------------------------------------------------------------------
__global__ void zero_kernel(float* __restrict__ p, int n) {
  int i = blockIdx.x * blockDim.x + threadIdx.x;
  int stride = gridDim.x * blockDim.x;
  for (; i < n; i += stride) p[i] = 0.0f;
}

__global__ void deg_kernel(const int* __restrict__ col, float* __restrict__ deg, int E) {
  int i = blockIdx.x * blockDim.x + threadIdx.x;
  if (i < E) atomicAdd(&deg[col[i]], 1.0f);
}

__global__ void dis_kernel(float* __restrict__ deg, int N) {
  int i = blockIdx.x * blockDim.x + threadIdx.x;
  if (i < N) {
    float d = deg[i];
    deg[i] = (d > 0.0f) ? rsqrtf(d) : 0.0f;
  }
}

__global__ void norm_kernel(const int* __restrict__ row, const int* __restrict__ col,
                            const float* __restrict__ dis, float* __restrict__ nrm, int E) {
  int i = blockIdx.x * blockDim.x + threadIdx.x;
  if (i < E) nrm[i] = dis[row[i]] * dis[col[i]];
}

// Combined per-layer edge weights: C_i = Wg @ eW_i ; d_i = bg @ eW_i + eb_i.
// Tiny (3 x 64x64x64 MACs); one block per layer.
__global__ void combine_kernel(const float* __restrict__ Wg, const float* __restrict__ bg,
                               const float* __restrict__ eW0, const float* __restrict__ eb0,
                               const float* __restrict__ eW1, const float* __restrict__ eb1,
                               const float* __restrict__ eW2, const float* __restrict__ eb2,
                               float* __restrict__ C, float* __restrict__ dv) {
  const float* eW = (blockIdx.x == 0) ? eW0 : (blockIdx.x == 1) ? eW1 : eW2;
  const float* eb = (blockIdx.x == 0) ? eb0 : (blockIdx.x == 1) ? eb1 : eb2;
  float* Co = C + (size_t)blockIdx.x * 64 * 64;
  float* dvo = dv + (size_t)blockIdx.x * 64;
  for (int idx = threadIdx.x; idx < 64 * 64; idx += blockDim.x) {
    int k = idx >> 6, n = idx & 63;
    float s = 0.0f;
#pragma unroll 8
    for (int j = 0; j < 64; ++j) s += Wg[k * 64 + j] * eW[j * 64 + n];
    Co[idx] = s;
  }
  if (threadIdx.x < 64) {
    int n = threadIdx.x;
    float s = eb[n];
#pragma unroll 8
    for (int j = 0; j < 64; ++j) s += bg[j] * eW[j * 64 + n];
    dvo[n] = s;
  }
}

// ---------------------------------------------------------------------------
// Node transform: out[N,64] = h[N,K] @ W[K,64] + bias.  One wave per 16-row
// tile.  A layout (32-bit 16x4): lane%16 = M, (lane/16,vgpr) = K.
// B layout (4x16): lane%16 = N, (lane/16,vgpr) = K.  C/D: lane%16 = N,
// vgpr j -> M = j + 8*(lane/16).
template <int K>
__global__ void __launch_bounds__(256) node_gemm_kernel(const float* __restrict__ h,
                                                        const float* __restrict__ W,
                                                        const float* __restrict__ bias,
                                                        float* __restrict__ out, int N) {
  const int lane = threadIdx.x & 31;
  const int lm = lane & 15;   // M (for A) / N (for B,D)
  const int lh = lane >> 4;   // K-half select
  const int wave = (blockIdx.x * blockDim.x + threadIdx.x) >> 5;
  const int nwaves = (gridDim.x * blockDim.x) >> 5;
  const int ntiles = N >> 4;

  // loop-invariant: bias for this lane's 4 column slots
  float bn4[4];
#pragma unroll
  for (int nt = 0; nt < 4; ++nt) bn4[nt] = bias[nt * 16 + lm];

  for (int t = wave; t < ntiles; t += nwaves) {
    const int base = t << 4;
    v8f accv[4] = {v8f{}, v8f{}, v8f{}, v8f{}};
    const float* arow = h + (size_t)(base + lm) * K + 2 * lh;
#pragma unroll
    for (int kc = 0; kc < (K >> 2); ++kc) {
      v2f a = *(const v2f*)(arow + kc * 4);               // K = 4kc+2lh, +1
      const float* bp = W + (size_t)(kc * 4 + 2 * lh) * 64 + lm;
#pragma unroll
      for (int nt = 0; nt < 4; ++nt) {
        v2f b = {bp[nt * 16], bp[64 + nt * 16]};
        accv[nt] = wmma4(a, b, accv[nt]);
      }
    }
    // store: one base pointer per output row, 4 stores at immediate offsets
#pragma unroll
    for (int j = 0; j < 8; ++j) {
      float* op = out + (size_t)(base + j + 8 * lh) * 64 + lm;
#pragma unroll
      for (int nt = 0; nt < 4; ++nt) op[nt * 16] = accv[nt][j] + bn4[nt];
    }
  }
}

// ---------------------------------------------------------------------------
// Fused edge kernel: for a 16-edge tile,
//   T = edge_attr[tile] @ C (WMMA, K=64)  ->  16x64 in registers
//   acc[col[e], n] += norm[e] * xw[row[e], n] * (T[e][n] + d[n])
__global__ void __launch_bounds__(256) edge_msg_kernel(const float* __restrict__ ea,
                                                       const float* __restrict__ C,
                                                       const float* __restrict__ dvec,
                                                       const int* __restrict__ row,
                                                       const int* __restrict__ col,
                                                       const float* __restrict__ nrm,
                                                       const float* __restrict__ xw,
                                                       float* __restrict__ acc, int E) {
  const int lane = threadIdx.x & 31;
  const int lm = lane & 15;
  const int lh = lane >> 4;
  const int wave = (blockIdx.x * blockDim.x + threadIdx.x) >> 5;
  const int nwaves = (gridDim.x * blockDim.x) >> 5;
  const int ntiles = E >> 4;

  // loop-invariant: combined bias for this lane's 4 column slots
  float dv4[4];
#pragma unroll
  for (int nt = 0; nt < 4; ++nt) dv4[nt] = dvec[nt * 16 + lm];

  for (int t = wave; t < ntiles; t += nwaves) {
    const int base = t << 4;
    v8f accv[4] = {v8f{}, v8f{}, v8f{}, v8f{}};
    const float* arow = ea + (size_t)(base + lm) * 64 + 2 * lh;
#pragma unroll
    for (int kc = 0; kc < 16; ++kc) {
      v2f a = *(const v2f*)(arow + kc * 4);
      const float* bp = C + (size_t)(kc * 4 + 2 * lh) * 64 + lm;
#pragma unroll
      for (int nt = 0; nt < 4; ++nt) {
        v2f b = {bp[nt * 16], bp[64 + nt * 16]};
        accv[nt] = wmma4(a, b, accv[nt]);
      }
    }

    // per-lane edge metadata for the 8 rows this lane's D registers cover
    int r8[8], c8[8];
    float nm8[8];
#pragma unroll
    for (int j = 0; j < 8; ++j) {
      const int e = base + j + 8 * lh;
      r8[j] = row[e];
      c8[j] = col[e];
      nm8[j] = nrm[e];
    }

    // j-outer / nt-inner: one base address per gather row and per scatter
    // row; the 4 column tiles become immediate offsets (0/64/128/192 B).
#pragma unroll
    for (int j = 0; j < 8; ++j) {
      const float* gp = xw + (size_t)r8[j] * 64 + lm;   // gather row base
      float* ap = acc + (size_t)c8[j] * 64 + lm;        // scatter row base
      const float nm = nm8[j];
#pragma unroll
      for (int nt = 0; nt < 4; ++nt) {
        const float eft = accv[nt][j] + dv4[nt];        // ef @ eW + eb
        atomicAdd(ap + nt * 16, nm * gp[nt * 16] * eft);
      }
    }
  }
}

// ---------------------------------------------------------------------------
// ReLU + LayerNorm over dim 64: one wave per row, 2 elems per lane.
__global__ void __launch_bounds__(256) relu_ln_kernel(const float* __restrict__ in,
                                                      const float* __restrict__ g,
                                                      const float* __restrict__ bt,
                                                      float* __restrict__ out, int N) {
  const int lane = threadIdx.x & 31;
  const int wave = (blockIdx.x * blockDim.x + threadIdx.x) >> 5;
  const int nwaves = (gridDim.x * blockDim.x) >> 5;
  for (int r = wave; r < N; r += nwaves) {
    v2f v = *(const v2f*)(in + (size_t)r * 64 + lane * 2);
    v.x = fmaxf(v.x, 0.0f);
    v.y = fmaxf(v.y, 0.0f);
    float s = v.x + v.y;
#pragma unroll
    for (int m = 16; m >= 1; m >>= 1) s += __shfl_xor(s, m, 32);
    const float mu = s * (1.0f / 64.0f);
    const float dx = v.x - mu, dy = v.y - mu;
    float sq = dx * dx + dy * dy;
#pragma unroll
    for (int m = 16; m >= 1; m >>= 1) sq += __shfl_xor(sq, m, 32);
    const float inv = rsqrtf(sq * (1.0f / 64.0f) + 1e-5f);
    v2f gg = *(const v2f*)(g + lane * 2);
    v2f bb = *(const v2f*)(bt + lane * 2);
    v2f o;
    o.x = dx * inv * gg.x + bb.x;
    o.y = dy * inv * gg.y + bb.y;
    *(v2f*)(out + (size_t)r * 64 + lane * 2) = o;
  }
}

// ---------------------------------------------------------------------------
extern "C" void kernel_launch(void* const* d_in, const int* in_sizes, int n_in,
                              void* d_out, int out_size, void* d_ws, size_t ws_size,
                              hipStream_t stream) {
  const float* x   = (const float*)d_in[0];
  const int*   ei  = (const int*)d_in[1];
  const float* ea  = (const float*)d_in[2];
  const float* Wg  = (const float*)d_in[3];
  const float* bg  = (const float*)d_in[4];
  const float* W0  = (const float*)d_in[5];
  const float* b0  = (const float*)d_in[6];
  const float* eW0 = (const float*)d_in[7];
  const float* eb0 = (const float*)d_in[8];
  const float* W1  = (const float*)d_in[9];
  const float* b1  = (const float*)d_in[10];
  const float* eW1 = (const float*)d_in[11];
  const float* eb1 = (const float*)d_in[12];
  const float* W2  = (const float*)d_in[13];
  const float* b2  = (const float*)d_in[14];
  const float* eW2 = (const float*)d_in[15];
  const float* eb2 = (const float*)d_in[16];
  const float* g0  = (const float*)d_in[17];
  const float* bt0 = (const float*)d_in[18];
  const float* g1  = (const float*)d_in[19];
  const float* bt1 = (const float*)d_in[20];

  const int N = in_sizes[0] / 128;   // 50000
  const int E = in_sizes[1] / 2;     // 800000
  const int* row = ei;
  const int* col = ei + E;

  // workspace carve-out (≈42 MB total)
  char* wsb = (char*)d_ws;
  size_t off = 0;
  auto carve = [&](size_t bytes) -> void* {
    void* p = wsb + off;
    off = (off + bytes + 255) & ~(size_t)255;
    return p;
  };
  float* dis  = (float*)carve((size_t)N * 4);             // deg -> deg^-0.5
  float* nrm  = (float*)carve((size_t)E * 4);             // per-edge norm
  float* xw   = (float*)carve((size_t)N * 64 * 4);        // h @ W + b
  float* hbuf = (float*)carve((size_t)N * 64 * 4);        // layer output
  float* acc  = (float*)carve((size_t)N * 64 * 4);        // scatter accumulator
  float* Cw   = (float*)carve((size_t)3 * 64 * 64 * 4);   // combined edge weights
  float* dv   = (float*)carve((size_t)3 * 64 * 4);        // combined edge biases
  (void)ws_size; (void)n_in; (void)out_size;

  float* out = (float*)d_out;
  const int TB = 256;
  const int eBlocks = (E + TB - 1) / TB;
  const int nBlocks = (N + TB - 1) / TB;
  const int gemmBlocksN = ((N >> 4) + 7) / 8;   // 8 waves / block, 1 tile / wave
  const int gemmBlocksE = ((E >> 4) + 7) / 8;
  const int lnBlocks = (N + 7) / 8;

  // degree normalization (shared by all layers)
  zero_kernel<<<nBlocks, TB, 0, stream>>>(dis, N);
  deg_kernel<<<eBlocks, TB, 0, stream>>>(col, dis, E);
  dis_kernel<<<nBlocks, TB, 0, stream>>>(dis, N);
  norm_kernel<<<eBlocks, TB, 0, stream>>>(row, col, dis, nrm, E);

  // fold global edge transform into each layer's edge weights
  combine_kernel<<<3, TB, 0, stream>>>(Wg, bg, eW0, eb0, eW1, eb1, eW2, eb2, Cw, dv);

  // ---- layer 0 ----
  node_gemm_kernel<128><<<gemmBlocksN, TB, 0, stream>>>(x, W0, b0, xw, N);
  zero_kernel<<<8192, TB, 0, stream>>>(acc, N * 64);
  edge_msg_kernel<<<gemmBlocksE, TB, 0, stream>>>(ea, Cw, dv, row, col, nrm, xw, acc, E);
  relu_ln_kernel<<<lnBlocks, TB, 0, stream>>>(acc, g0, bt0, hbuf, N);

  // ---- layer 1 ----
  node_gemm_kernel<64><<<gemmBlocksN, TB, 0, stream>>>(hbuf, W1, b1, xw, N);
  zero_kernel<<<8192, TB, 0, stream>>>(acc, N * 64);
  edge_msg_kernel<<<gemmBlocksE, TB, 0, stream>>>(ea, Cw + 4096, dv + 64, row, col, nrm, xw, acc, E);
  relu_ln_kernel<<<lnBlocks, TB, 0, stream>>>(acc, g1, bt1, hbuf, N);

  // ---- layer 2 (accumulate straight into d_out) ----
  node_gemm_kernel<64><<<gemmBlocksN, TB, 0, stream>>>(hbuf, W2, b2, xw, N);
  zero_kernel<<<8192, TB, 0, stream>>>(out, N * 64);
  edge_msg_kernel<<<gemmBlocksE, TB, 0, stream>>>(ea, Cw + 8192, dv + 128, row, col, nrm, xw, out, E);
}